// AtomAttentionEncoder_47502338294328
// MI455X (gfx1250) — compile-verified
//
#include <hip/hip_runtime.h>
#include <math.h>

// ---------------------------------------------------------------------------
// AtomAttentionEncoder forward for MI455X (gfx1250), fp32 end-to-end.
// All GEMMs + attention QK^T / PV use V_WMMA_F32_16X16X4_F32 (wave32 WMMA).
// Attention exploits the block-bias structure: each query row a attends only
// to the 128-key window [32*(a/32)-48, 32*(a/32)+79]; outside keys have
// logit -1e10 -> exp == 0.0f exactly in fp32, so skipping them is exact.
// plm [1536,1536,16] is written directly into its d_out slot and re-read by
// the attention kernels for the per-block pair bias LN(plm)@Wzb.
//
// Input flattening assumption: d_in follows setup_inputs() insertion order
// for the 11 top-level entries; 'params' is flattened JAX-style (dict keys
// sorted, lists in order):
//   0 ref_pos 1 ref_mask 2 ref_element 3 ref_charge 4 ref_atom_name_chars
//   5 ref_space_uid 6 rl 7 si_trunk 8 zij 9 atom_mask 10 atom_tok_idx(int32)
//   11.. params: W_feats,W_inv,W_l,W_m,W_mlp1,W_mlp2,W_mlp3,W_off,W_qtok,
//                W_r,W_s,W_vm,W_z, blocks[0..2]{Wg2,Wk,Wo,Wog,Wq,Wt1,Wt2,Wt3,
//                Wtg,Wv,Wzb, ada{Wb,Wg,b,bg,g}, ada2{Wb,Wg,b,bg,g},
//                bog,bq,btg,lnz_b,lnz_g}, lnS_b,lnS_g,lnZ_b,lnZ_g
// d_out: ai[384*384] | a[1536*128] | cl[1536*128] | plm[1536*1536*16]
// ---------------------------------------------------------------------------

#define NA 1536
#define NT 384
#define CA 128
#define CP 16
#define CS 384
#define CZ 128
#define NHEAD 4
#define CHD 32

typedef float v2f __attribute__((ext_vector_type(2)));
typedef float v8f __attribute__((ext_vector_type(8)));

__device__ __forceinline__ float sigm(float x) { return 1.0f / (1.0f + __expf(-x)); }

__device__ __forceinline__ v8f wmma4(v2f a, v2f b, v8f c) {
  // D = A(16x4) * B(4x16) + C(16x16), fp32
  return __builtin_amdgcn_wmma_f32_16x16x4_f32(false, a, false, b, (short)0, c, false, false);
}

// --------------------------- generic WMMA GEMM -----------------------------
// C[M,N] = act(A[M,K] @ W[K,N] + bias).  One wave per 16x16 tile, 4 waves/WG.
// Fragment layouts per CDNA5 ISA 7.12.2:
//   A: lane l holds A[l&15][2*(l>>4)+j] in vgpr j
//   B: lane l holds B[2*(l>>4)+j][l&15] in vgpr j
//   C: lane l holds C[r+8*(l>>4)][l&15] in vgpr r
__global__ __launch_bounds__(128) void gemm_k(const float* __restrict__ A,
                                              const float* __restrict__ W,
                                              const float* __restrict__ bias,
                                              float* __restrict__ C,
                                              int M, int N, int K, int act) {
  const int lane = threadIdx.x & 31;
  const int wave = threadIdx.x >> 5;
  const int half = lane >> 4;
  const int n    = lane & 15;
  const int m0   = (blockIdx.y * 4 + wave) * 16;
  const int n0   = blockIdx.x * 16;
  if (m0 >= M) return;  // uniform per wave
  v8f acc = {0.f, 0.f, 0.f, 0.f, 0.f, 0.f, 0.f, 0.f};
  const float* arow = A + (size_t)(m0 + n) * K;
  for (int k = 0; k < K; k += 4) {
    const int ka = k + 2 * half;
    v2f av; av.x = arow[ka];                     av.y = arow[ka + 1];
    v2f bv; bv.x = W[(size_t)ka * N + n0 + n];   bv.y = W[(size_t)(ka + 1) * N + n0 + n];
    acc = wmma4(av, bv, acc);
  }
  const float bval = bias ? bias[n0 + n] : 0.0f;
#pragma unroll
  for (int r = 0; r < 8; ++r) {
    float v = acc[r] + bval;
    if (act == 1) v = fmaxf(v, 0.0f);
    C[(size_t)(m0 + r + 8 * half) * N + n0 + n] = v;
  }
}

// --------------------------- elementwise kernel ----------------------------
// mode 0: out = sigmoid(x)*y + z   mode 1: out = sigmoid(x)*y
// mode 2: out = silu(x)*y          mode 4: out = relu(x)   mode 5: out = x
__global__ void ew_k(const float* __restrict__ x, const float* __restrict__ y,
                     const float* __restrict__ z, float* __restrict__ out,
                     int n, int mode) {
  int i = blockIdx.x * blockDim.x + threadIdx.x;
  if (i >= n) return;
  float r;
  switch (mode) {
    case 0: r = z[i] + y[i] * sigm(x[i]); break;
    case 1: r = y[i] * sigm(x[i]); break;
    case 2: r = x[i] * sigm(x[i]) * y[i]; break;
    case 4: r = fmaxf(x[i], 0.0f); break;
    default: r = x[i]; break;
  }
  out[i] = r;
}

// ----------------------- LayerNorm of si_trunk rows ------------------------
__global__ void siln_k(const float* __restrict__ si, const float* __restrict__ g,
                       const float* __restrict__ b, float* __restrict__ out) {
  int t = blockIdx.x * blockDim.x + threadIdx.x;
  if (t >= NT) return;
  const float* r = si + (size_t)t * CS;
  float mu = 0.f;
  for (int j = 0; j < CS; ++j) mu += r[j];
  mu *= (1.0f / CS);
  float var = 0.f;
  for (int j = 0; j < CS; ++j) { float d = r[j] - mu; var += d * d; }
  float rs = rsqrtf(var * (1.0f / CS) + 1e-5f);
  float* o = out + (size_t)t * CS;
  for (int j = 0; j < CS; ++j) o[j] = (r[j] - mu) * rs * g[j] + b[j];
}

// ------------- cl = feats @ W_feats (+ gathered LN(si) @ W_s) --------------
__global__ void cl_k(const float* __restrict__ pos, const float* __restrict__ msk,
                     const float* __restrict__ elem, const float* __restrict__ chg,
                     const float* __restrict__ chars, const float* __restrict__ uid,
                     const float* __restrict__ Wf, const float* __restrict__ siLN,
                     const float* __restrict__ Ws, const int* __restrict__ tok,
                     float* __restrict__ pre, float* __restrict__ cl_out) {
  int i = blockIdx.x * blockDim.x + threadIdx.x;
  if (i >= NA * CA) return;
  const int a = i >> 7, c = i & 127;
  const float* W = Wf + c;  // column c, row stride CA
  float acc = pos[a * 3 + 0] * W[0 * CA] + pos[a * 3 + 1] * W[1 * CA] +
              pos[a * 3 + 2] * W[2 * CA] + msk[a] * W[3 * CA];
  const float* el = elem + (size_t)a * 128;
  for (int j = 0; j < 128; ++j) acc += el[j] * W[(size_t)(4 + j) * CA];
  acc += chg[a] * W[132 * CA];
  const float* ch = chars + (size_t)a * 256;
  for (int j = 0; j < 256; ++j) acc += ch[j] * W[(size_t)(133 + j) * CA];
  acc += uid[a] * W[389 * CA];
  pre[i] = acc;  // ql base (stop_gradient copy, before si term)
  const float* sr = siLN + (size_t)tok[a] * CS;
  float s2 = 0.f;
  for (int j = 0; j < CS; ++j) s2 += sr[j] * Ws[(size_t)j * CA + c];
  cl_out[i] = acc + s2;
}

// ----------------------- ql = pre + rl @ W_r -------------------------------
__global__ void ql_k(const float* __restrict__ pre, const float* __restrict__ rl,
                     const float* __restrict__ Wr, float* __restrict__ acur) {
  int i = blockIdx.x * blockDim.x + threadIdx.x;
  if (i >= NA * CA) return;
  const int a = i >> 7, c = i & 127;
  acur[i] = pre[i] + rl[a * 3 + 0] * Wr[c] + rl[a * 3 + 1] * Wr[CA + c] +
            rl[a * 3 + 2] * Wr[2 * CA + c];
}

// ------------------- zproj = LN(zij) @ W_z  [T,T,16] -----------------------
__global__ void zproj_k(const float* __restrict__ zij, const float* __restrict__ g,
                        const float* __restrict__ b, const float* __restrict__ Wz,
                        float* __restrict__ out) {
  int i = blockIdx.x * blockDim.x + threadIdx.x;
  if (i >= NT * NT) return;
  const float* r = zij + (size_t)i * CZ;
  float mu = 0.f;
  for (int j = 0; j < CZ; ++j) mu += r[j];
  mu *= (1.0f / CZ);
  float var = 0.f;
  for (int j = 0; j < CZ; ++j) { float d = r[j] - mu; var += d * d; }
  float rs = rsqrtf(var * (1.0f / CZ) + 1e-5f);
  float acc[CP];
#pragma unroll
  for (int c = 0; c < CP; ++c) acc[c] = 0.f;
  for (int j = 0; j < CZ; ++j) {
    float ln = (r[j] - mu) * rs * g[j] + b[j];
#pragma unroll
    for (int c = 0; c < CP; ++c) acc[c] += ln * Wz[j * CP + c];
  }
  float* o = out + (size_t)i * CP;
#pragma unroll
  for (int c = 0; c < CP; ++c) o[c] = acc[c];
}

// --------------------- plm pair rep + 16x16 MLP ----------------------------
__global__ void plm_k(const float* __restrict__ pos, const float* __restrict__ uid,
                      const int* __restrict__ tok, const float* __restrict__ zproj,
                      const float* __restrict__ rcWl, const float* __restrict__ rcWm,
                      const float* __restrict__ Woff, const float* __restrict__ Winv,
                      const float* __restrict__ Wvm, const float* __restrict__ W1,
                      const float* __restrict__ W2, const float* __restrict__ W3,
                      float* __restrict__ plm) {
  size_t i = (size_t)blockIdx.x * blockDim.x + threadIdx.x;
  if (i >= (size_t)NA * NA) return;
  const int a = (int)(i / NA), b = (int)(i % NA);
  const float d0 = pos[b * 3 + 0] - pos[a * 3 + 0];
  const float d1 = pos[b * 3 + 1] - pos[a * 3 + 1];
  const float d2 = pos[b * 3 + 2] - pos[a * 3 + 2];
  const float v  = (uid[a] == uid[b]) ? 1.0f : 0.0f;
  const float sq = d0 * d0 + d1 * d1 + d2 * d2;
  const float dist = sq > 0.0f ? sqrtf(sq) : 0.0f;
  const float invd = 1.0f / (1.0f + dist);
  const float* zp = zproj + ((size_t)tok[a] * NT + tok[b]) * CP;
  const float* wl = rcWl + (size_t)b * CP;
  const float* wm = rcWm + (size_t)a * CP;
  float pv[CP], h1[CP], h2[CP];
#pragma unroll
  for (int c = 0; c < CP; ++c)
    pv[c] = v * (d0 * Woff[c] + d1 * Woff[CP + c] + d2 * Woff[2 * CP + c] +
                 invd * Winv[c] + Wvm[c]) +
            zp[c] + wl[c] + wm[c];
#pragma unroll
  for (int c = 0; c < CP; ++c) {
    float acc = 0.f;
#pragma unroll
    for (int j = 0; j < CP; ++j) acc += fmaxf(pv[j], 0.f) * W1[j * CP + c];
    h1[c] = acc;
  }
#pragma unroll
  for (int c = 0; c < CP; ++c) {
    float acc = 0.f;
#pragma unroll
    for (int j = 0; j < CP; ++j) acc += fmaxf(h1[j], 0.f) * W2[j * CP + c];
    h2[c] = acc;
  }
  float* po = plm + i * CP;
#pragma unroll
  for (int c = 0; c < CP; ++c) {
    float acc = 0.f;
#pragma unroll
    for (int j = 0; j < CP; ++j) acc += fmaxf(h2[j], 0.f) * W3[j * CP + c];
    po[c] = pv[c] + acc;
  }
}

// ----------- an = LN(a) (no affine); sn = LN(cl)*g + b  per row ------------
__global__ void ln2_k(const float* __restrict__ a, const float* __restrict__ cl,
                      const float* __restrict__ g, const float* __restrict__ b,
                      float* __restrict__ an, float* __restrict__ sn) {
  int i = blockIdx.x * blockDim.x + threadIdx.x;
  if (i >= NA) return;
  {
    const float* r = a + (size_t)i * CA;
    float mu = 0.f;
    for (int j = 0; j < CA; ++j) mu += r[j];
    mu *= (1.0f / CA);
    float var = 0.f;
    for (int j = 0; j < CA; ++j) { float d = r[j] - mu; var += d * d; }
    float rs = rsqrtf(var * (1.0f / CA) + 1e-5f);
    float* o = an + (size_t)i * CA;
    for (int j = 0; j < CA; ++j) o[j] = (r[j] - mu) * rs;
  }
  {
    const float* r = cl + (size_t)i * CA;
    float mu = 0.f;
    for (int j = 0; j < CA; ++j) mu += r[j];
    mu *= (1.0f / CA);
    float var = 0.f;
    for (int j = 0; j < CA; ++j) { float d = r[j] - mu; var += d * d; }
    float rs = rsqrtf(var * (1.0f / CA) + 1e-5f);
    float* o = sn + (size_t)i * CA;
    for (int j = 0; j < CA; ++j) o[j] = (r[j] - mu) * rs * g[j] + b[j];
  }
}

// ------------------- fused local pair-biased attention ---------------------
// 4 waves/WG: wave h handles head h of a 16-row tile; 128-key local window.
__global__ __launch_bounds__(128) void attn_k(
    const float* __restrict__ qb, const float* __restrict__ kb,
    const float* __restrict__ vb, const float* __restrict__ g2b,
    const float* __restrict__ plm, const float* __restrict__ lnzg,
    const float* __restrict__ lnzb, const float* __restrict__ Wzb,
    const float* __restrict__ amask, float* __restrict__ obuf) {
  __shared__ float sz[NHEAD][16][128];  // 32 KB: zbias, then S, then P
  const int tid = threadIdx.x;
  const int a0  = blockIdx.x * 16;
  const int w0  = (a0 & ~31) - 48;  // window start for this 32-row block
  // Phase 1 (cooperative): zb[h][m][bc] = LN(plm[a0+m, w0+bc,:]) @ Wzb[:,h] + mask bias
  for (int idx = tid; idx < 16 * 128; idx += 128) {
    const int m = idx >> 7, bc = idx & 127;
    const int col = w0 + bc;
    float o0 = -1e30f, o1 = -1e30f, o2 = -1e30f, o3 = -1e30f;
    if (col >= 0 && col < NA) {
      const float* pr = plm + ((size_t)(a0 + m) * NA + col) * CP;
      float x[CP];
      float mu = 0.f;
#pragma unroll
      for (int c = 0; c < CP; ++c) { x[c] = pr[c]; mu += x[c]; }
      mu *= (1.0f / CP);
      float var = 0.f;
#pragma unroll
      for (int c = 0; c < CP; ++c) { float d = x[c] - mu; var += d * d; }
      float rs = rsqrtf(var * (1.0f / CP) + 1e-5f);
      float mb = (amask[col] - 1.0f) * 1e9f;
      o0 = mb; o1 = mb; o2 = mb; o3 = mb;
#pragma unroll
      for (int c = 0; c < CP; ++c) {
        float ln = (x[c] - mu) * rs * lnzg[c] + lnzb[c];
        o0 += ln * Wzb[c * 4 + 0];
        o1 += ln * Wzb[c * 4 + 1];
        o2 += ln * Wzb[c * 4 + 2];
        o3 += ln * Wzb[c * 4 + 3];
      }
    }
    sz[0][m][bc] = o0; sz[1][m][bc] = o1; sz[2][m][bc] = o2; sz[3][m][bc] = o3;
  }
  __syncthreads();
  const int h    = tid >> 5;
  const int lane = tid & 31;
  const int half = lane >> 4;
  const int n    = lane & 15;
  const float scale = 0.17677669529663687f;  // 1/sqrt(32)
  const float* qrow = qb + (size_t)(a0 + n) * CA + h * CHD;
  // Phase 2: S tiles via chained WMMA (K=32 -> 8 steps of 4)
  for (int t = 0; t < 8; ++t) {
    v8f acc = {0.f, 0.f, 0.f, 0.f, 0.f, 0.f, 0.f, 0.f};
    int kcol = w0 + t * 16 + n;
    int kc = kcol < 0 ? 0 : (kcol >= NA ? NA - 1 : kcol);  // clamped; p==0 there
    const float* krow = kb + (size_t)kc * CA + h * CHD;
#pragma unroll
    for (int kk = 0; kk < 8; ++kk) {
      const int ch = kk * 4 + 2 * half;
      v2f av; av.x = qrow[ch]; av.y = qrow[ch + 1];
      v2f bv; bv.x = krow[ch]; bv.y = krow[ch + 1];
      acc = wmma4(av, bv, acc);
    }
#pragma unroll
    for (int r = 0; r < 8; ++r) {
      const int m = r + 8 * half;
      sz[h][m][t * 16 + n] = acc[r] * scale + sz[h][m][t * 16 + n];
    }
  }
  // Phase 3: softmax over the 128-window (2 lanes per row)
  {
    const int m = n;
    const int c0 = half * 64;
    float mx = -3.0e38f;
    for (int j = 0; j < 64; ++j) mx = fmaxf(mx, sz[h][m][c0 + j]);
    mx = fmaxf(mx, __shfl_xor(mx, 16));
    float sum = 0.f;
    for (int j = 0; j < 64; ++j) {
      float e = __expf(sz[h][m][c0 + j] - mx);
      sz[h][m][c0 + j] = e;
      sum += e;
    }
    sum += __shfl_xor(sum, 16);
    const float inv = 1.0f / sum;
    for (int j = 0; j < 64; ++j) sz[h][m][c0 + j] *= inv;
  }
  // Phase 4: O = P(16x128) @ V(128x32), gated by sigmoid(g2)
  for (int chh = 0; chh < 2; ++chh) {
    v8f acc = {0.f, 0.f, 0.f, 0.f, 0.f, 0.f, 0.f, 0.f};
    for (int g = 0; g < 32; ++g) {
      const int kbase = 4 * g + 2 * half;
      v2f av; av.x = sz[h][n][kbase]; av.y = sz[h][n][kbase + 1];
      int c0i = w0 + kbase;
      int cc0 = c0i < 0 ? 0 : (c0i >= NA ? NA - 1 : c0i);
      int cc1 = (c0i + 1) < 0 ? 0 : ((c0i + 1) >= NA ? NA - 1 : c0i + 1);
      const int ch = h * CHD + chh * 16 + n;
      v2f bv; bv.x = vb[(size_t)cc0 * CA + ch]; bv.y = vb[(size_t)cc1 * CA + ch];
      acc = wmma4(av, bv, acc);
    }
#pragma unroll
    for (int r = 0; r < 8; ++r) {
      const int row = a0 + r + 8 * half;
      const int ch = h * CHD + chh * 16 + n;
      obuf[(size_t)row * CA + ch] = sigm(g2b[(size_t)row * CA + ch]) * acc[r];
    }
  }
}

// ------------------- token mean of relu(a @ W_qtok) ------------------------
__global__ void segmean_k(const float* __restrict__ qtok, float* __restrict__ ai) {
  int i = blockIdx.x * blockDim.x + threadIdx.x;
  if (i >= NT * NT) return;
  const int t = i / NT, c = i % NT;
  float s = 0.f;
  for (int j = 0; j < 4; ++j) s += qtok[(size_t)(t * 4 + j) * NT + c];
  ai[i] = s * 0.25f;
}

// ---------------------------------------------------------------------------
extern "C" void kernel_launch(void* const* d_in, const int* in_sizes, int n_in,
                              void* d_out, int out_size, void* d_ws, size_t ws_size,
                              hipStream_t stream) {
  (void)in_sizes; (void)n_in; (void)out_size; (void)ws_size;
  auto F = [&](int i) { return (const float*)d_in[i]; };
  const float* ref_pos    = F(0);
  const float* ref_mask   = F(1);
  const float* ref_elem   = F(2);
  const float* ref_charge = F(3);
  const float* ref_chars  = F(4);
  const float* ref_uid    = F(5);
  const float* rl         = F(6);
  const float* si_trunk   = F(7);
  const float* zij        = F(8);
  const float* atom_mask  = F(9);
  const int*   tok        = (const int*)d_in[10];
  int p = 11;  // params leaves, JAX-flatten order (sorted keys)
  const float* W_feats = F(p++); const float* W_inv = F(p++);
  const float* W_l = F(p++);     const float* W_m = F(p++);
  const float* W_mlp1 = F(p++);  const float* W_mlp2 = F(p++); const float* W_mlp3 = F(p++);
  const float* W_off = F(p++);   const float* W_qtok = F(p++); const float* W_r = F(p++);
  const float* W_s = F(p++);     const float* W_vm = F(p++);   const float* W_z = F(p++);
  struct Blk {
    const float *Wg2, *Wk, *Wo, *Wog, *Wq, *Wt1, *Wt2, *Wt3, *Wtg, *Wv, *Wzb;
    const float *aWb, *aWg, *ab, *abg, *ag;
    const float *a2Wb, *a2Wg, *a2b, *a2bg, *a2g;
    const float *bog, *bq, *btg, *lnz_b, *lnz_g;
  } blk[3];
  for (int b = 0; b < 3; ++b) {
    Blk& B = blk[b];
    B.Wg2 = F(p++); B.Wk = F(p++); B.Wo = F(p++); B.Wog = F(p++); B.Wq = F(p++);
    B.Wt1 = F(p++); B.Wt2 = F(p++); B.Wt3 = F(p++); B.Wtg = F(p++); B.Wv = F(p++);
    B.Wzb = F(p++);
    B.aWb = F(p++); B.aWg = F(p++); B.ab = F(p++); B.abg = F(p++); B.ag = F(p++);
    B.a2Wb = F(p++); B.a2Wg = F(p++); B.a2b = F(p++); B.a2bg = F(p++); B.a2g = F(p++);
    B.bog = F(p++); B.bq = F(p++); B.btg = F(p++); B.lnz_b = F(p++); B.lnz_g = F(p++);
  }
  const float* lnS_b = F(p++); const float* lnS_g = F(p++);
  const float* lnZ_b = F(p++); const float* lnZ_g = F(p++);

  // workspace layout (floats) ~27.5 MB
  float* ws    = (float*)d_ws;
  float* siLN  = ws;                      // 384*384
  float* zproj = siLN + NT * CS;          // 384*384*16
  float* pre   = zproj + (size_t)NT * NT * CP;
  float* rcbuf = pre + NA * CA;
  float* rcWl  = rcbuf + NA * CA;
  float* rcWm  = rcWl + NA * CP;
  float* abuf0 = rcWm + NA * CP;
  float* abuf1 = abuf0 + NA * CA;
  float* anb   = abuf1 + NA * CA;
  float* snb   = anb + NA * CA;
  float* t1    = snb + NA * CA;           // NA*256
  float* t2    = t1 + NA * 256;           // NA*256
  float* alnb  = t2 + NA * 256;
  float* qbuf  = alnb + NA * CA;
  float* kbuf  = qbuf + NA * CA;
  float* vbuf  = kbuf + NA * CA;
  float* g2buf = vbuf + NA * CA;
  float* obuf  = g2buf + NA * CA;
  float* battn = obuf + NA * CA;
  float* hbuf  = battn + NA * CA;         // NA*256
  float* qtok  = hbuf + NA * 256;         // NA*384

  float* out     = (float*)d_out;
  float* out_ai  = out;                   // [384,384]
  float* out_a   = out + NT * NT;         // [1536,128]
  float* out_cl  = out_a + NA * CA;       // [1536,128]
  float* out_plm = out_cl + NA * CA;      // [1536,1536,16]

  auto gemm = [&](const float* A, const float* W, const float* bias, float* C,
                  int N, int K, int act) {
    dim3 g(N / 16, NA / 64);
    gemm_k<<<g, 128, 0, stream>>>(A, W, bias, C, NA, N, K, act);
  };
  auto ew = [&](const float* x, const float* y, const float* z, float* o, int n, int mode) {
    ew_k<<<(n + 255) / 256, 256, 0, stream>>>(x, y, z, o, n, mode);
  };

  // ---- feature / pair-rep stage ----
  siln_k<<<3, 128, 0, stream>>>(si_trunk, lnS_g, lnS_b, siLN);
  cl_k<<<(NA * CA + 255) / 256, 256, 0, stream>>>(ref_pos, ref_mask, ref_elem, ref_charge,
                                                  ref_chars, ref_uid, W_feats, siLN, W_s,
                                                  tok, pre, out_cl);
  ql_k<<<(NA * CA + 255) / 256, 256, 0, stream>>>(pre, rl, W_r, abuf0);
  ew(out_cl, nullptr, nullptr, rcbuf, NA * CA, 4);  // rc = relu(cl)
  zproj_k<<<(NT * NT + 127) / 128, 128, 0, stream>>>(zij, lnZ_g, lnZ_b, W_z, zproj);
  gemm(rcbuf, W_l, nullptr, rcWl, 16, 128, 0);
  gemm(rcbuf, W_m, nullptr, rcWm, 16, 128, 0);
  plm_k<<<(int)(((size_t)NA * NA + 255) / 256), 256, 0, stream>>>(
      ref_pos, ref_uid, tok, zproj, rcWl, rcWm, W_off, W_inv, W_vm,
      W_mlp1, W_mlp2, W_mlp3, out_plm);

  // ---- 3 transformer blocks ----
  float* acur = abuf0;
  float* anxt = abuf1;
  for (int b = 0; b < 3; ++b) {
    Blk& B = blk[b];
    // adaLN for attention
    ln2_k<<<(NA + 127) / 128, 128, 0, stream>>>(acur, out_cl, B.ag, B.ab, anb, snb);
    gemm(snb, B.aWg, B.abg, t1, 128, 128, 0);
    gemm(snb, B.aWb, nullptr, t2, 128, 128, 0);
    ew(t1, anb, t2, alnb, NA * CA, 0);  // aln = sigmoid(.)*an + sn@Wb
    // projections
    gemm(alnb, B.Wq, B.bq, qbuf, 128, 128, 0);
    gemm(alnb, B.Wk, nullptr, kbuf, 128, 128, 0);
    gemm(alnb, B.Wv, nullptr, vbuf, 128, 128, 0);
    gemm(alnb, B.Wg2, nullptr, g2buf, 128, 128, 0);
    // fused local attention (+ sigmoid(g2) gate)
    attn_k<<<NA / 16, 128, 0, stream>>>(qbuf, kbuf, vbuf, g2buf, out_plm,
                                        B.lnz_g, B.lnz_b, B.Wzb, atom_mask, obuf);
    gemm(obuf, B.Wo, nullptr, t1, 128, 128, 0);
    gemm(out_cl, B.Wog, B.bog, t2, 128, 128, 0);
    ew(t2, t1, nullptr, battn, NA * CA, 1);  // battn = sigmoid(cl@Wog+bog)*out
    // adaLN for transition
    ln2_k<<<(NA + 127) / 128, 128, 0, stream>>>(acur, out_cl, B.a2g, B.a2b, anb, snb);
    gemm(snb, B.a2Wg, B.a2bg, t1, 128, 128, 0);
    gemm(snb, B.a2Wb, nullptr, t2, 128, 128, 0);
    ew(t1, anb, t2, alnb, NA * CA, 0);
    // transition
    gemm(alnb, B.Wt1, nullptr, t1, 256, 128, 0);
    gemm(alnb, B.Wt2, nullptr, t2, 256, 128, 0);
    ew(t1, t2, nullptr, hbuf, NA * 256, 2);  // h = silu(u1)*u2
    gemm(hbuf, B.Wt3, nullptr, t1, 128, 256, 0);
    gemm(out_cl, B.Wtg, B.btg, t2, 128, 128, 0);
    ew(t2, t1, battn, anxt, NA * CA, 0);  // a = battn + sigmoid(cl@Wtg+btg)*(h@Wt3)
    float* tmp = acur; acur = anxt; anxt = tmp;
  }

  // ---- outputs ----
  gemm(acur, W_qtok, nullptr, qtok, 384, 128, 1);  // relu epilogue
  segmean_k<<<(NT * NT + 255) / 256, 256, 0, stream>>>(qtok, out_ai);
  ew(acur, nullptr, nullptr, out_a, NA * CA, 5);   // copy a
}